// CosineSSMLoss_5145370820963
// MI455X (gfx1250) — compile-verified
//
#include <hip/hip_runtime.h>
#include <hip/hip_bf16.h>
#include <math.h>

typedef __attribute__((ext_vector_type(2))) float v2f;
typedef __attribute__((ext_vector_type(8))) float v8f;

#define BATCH   4
#define CHAN    4
#define NPOS    4096          // 64*64
#define INV_ELEMS (2 * BATCH * NPOS)   // 32768 floats
#define NWAVES  32            // waves per batch-pair
#define NTILES  (2 * NWAVES)  // 64 partial 16x16 tiles
#define UNROLL  4             // independent WMMA accumulators per wave

// ---------------------------------------------------------------------------
// Kernel 1: per-position inverse channel-norm, matching F.normalize semantics:
//   inv[t,b,n] = 1 / max(sqrt(sum_c x[b,c,n]^2), 1e-12)
// ---------------------------------------------------------------------------
__global__ void cosssm_norm_kernel(const float* __restrict__ xp,
                                   const float* __restrict__ xs,
                                   float* __restrict__ inv) {
    int t = blockIdx.x * blockDim.x + threadIdx.x;   // 0 .. 32767
    if (t >= INV_ELEMS) return;
    int which = t >> 14;         // 0 = pred, 1 = src
    int b     = (t >> 12) & 3;
    int n     = t & (NPOS - 1);
    const float* x = which ? xs : xp;
    float ss = 0.0f;
#pragma unroll
    for (int c = 0; c < CHAN; ++c) {
        float v = x[(b * CHAN + c) * NPOS + n];
        ss += v * v;
    }
    float nm = sqrtf(ss);
    inv[t] = 1.0f / fmaxf(nm, 1e-12f);
}

// ---------------------------------------------------------------------------
// Kernel 2: WMMA channel-Gram accumulation.
// 16 rows of the WMMA A-matrix = [b0 z_pred c0..3 | b0 z_src c0..3 |
//                                 b1 z_pred c0..3 | b1 z_src c0..3]
// K = 4 consecutive spatial positions per V_WMMA_F32_16X16X4_F32; B operand
// is the value-transpose of A, which under the documented striped layouts is
// the same VGPR pair.  Four independent accumulators break the D->C serial
// chain so 4 WMMAs are in flight per outer iteration, with their 64-bit
// loads clustered ahead of them.
// ---------------------------------------------------------------------------
__global__ void __launch_bounds__(32)
cosssm_gram_kernel(const float* __restrict__ xp,
                   const float* __restrict__ xs,
                   const float* __restrict__ inv,
                   float* __restrict__ tiles) {
    const int pair  = blockIdx.x & 1;       // batch pair: (0,1) or (2,3)
    const int wslot = blockIdx.x >> 1;      // 0..31
    const int lane  = threadIdx.x;          // wave32

    const int m    = lane & 15;             // A-matrix row this lane feeds
    const int c    = m & 3;                 // channel
    const int half = (m >> 2) & 1;          // 0 = pred, 1 = src
    const int bsub = m >> 3;                // batch within pair
    const int b    = pair * 2 + bsub;

    const float* x    = half ? xs : xp;
    const float* row  = x   + (b * CHAN + c) * NPOS;
    const float* invw = inv + half * (BATCH * NPOS) + b * NPOS;

    // 32-bit A-matrix 16x4 layout: VGPR0 holds K=0 (lanes 0-15) / K=2
    // (lanes 16-31); VGPR1 holds K=1 / K=3.
    const int k0 = (lane >= 16) ? 2 : 0;

    v8f acc[UNROLL];
#pragma unroll
    for (int u = 0; u < UNROLL; ++u)
        acc[u] = (v8f){0.f, 0.f, 0.f, 0.f, 0.f, 0.f, 0.f, 0.f};

    // Each outer iteration covers UNROLL*4 = 16 consecutive positions.
    for (int n0 = wslot * (4 * UNROLL); n0 < NPOS; n0 += NWAVES * 4 * UNROLL) {
        v2f a[UNROLL];
#pragma unroll
        for (int u = 0; u < UNROLL; ++u) {
            int p0 = n0 + u * 4 + k0;      // p1 = p0 + 1 -> merged b64 loads
            a[u].x = row[p0]     * invw[p0];
            a[u].y = row[p0 + 1] * invw[p0 + 1];
        }
#pragma unroll
        for (int u = 0; u < UNROLL; ++u) {
            acc[u] = __builtin_amdgcn_wmma_f32_16x16x4_f32(
                /*neg_a=*/false, a[u], /*neg_b=*/false, a[u],
                /*c_mod=*/(short)0, acc[u], /*reuse_a=*/false, /*reuse_b=*/false);
        }
    }

    // Combine the independent accumulators (linear, deterministic order).
    v8f accT = (acc[0] + acc[1]) + (acc[2] + acc[3]);

    // C/D layout: VGPR v -> row v (lanes 0-15) or row v+8 (lanes 16-31),
    // column = lane & 15.  Store row-major 16x16 tile.
    float* tb = tiles + blockIdx.x * 256;
    const int rbase = (lane >= 16) ? 8 : 0;
    const int col   = lane & 15;
#pragma unroll
    for (int v = 0; v < 8; ++v)
        tb[(rbase + v) * 16 + col] = accT[v];
}

// ---------------------------------------------------------------------------
// Kernel 3: reduce partial tiles, combine Frobenius norms of the 4x4 blocks:
//   loss = sum_b ( |Gp_b|^2 + |Gs_b|^2 - 2|M_b|^2 ) / (B*N*N)
// where Gp = z_p z_p^T, Gs = z_s z_s^T, M = z_p z_s^T (all 4x4 per batch).
// ---------------------------------------------------------------------------
__global__ void cosssm_finalize_kernel(const float* __restrict__ tiles,
                                       float* __restrict__ out) {
    __shared__ float G[2 * 256];
    int tid  = threadIdx.x;          // blockDim.x == 512
    int pair = tid >> 8;
    int e    = tid & 255;
    float s = 0.0f;
    for (int w = 0; w < NWAVES; ++w)
        s += tiles[(w * 2 + pair) * 256 + e];
    G[pair * 256 + e] = s;
    __syncthreads();

    if (tid == 0) {
        float loss = 0.0f;
        for (int p = 0; p < 2; ++p) {
            const float* g = G + p * 256;
            for (int bs = 0; bs < 2; ++bs) {
                int r = bs * 8;
                float sp = 0.f, ssq = 0.f, sm = 0.f;
                for (int i = 0; i < 4; ++i) {
                    for (int j = 0; j < 4; ++j) {
                        float gp = g[(r + i) * 16 + (r + j)];         // z_p z_p^T
                        float gs = g[(r + 4 + i) * 16 + (r + 4 + j)]; // z_s z_s^T
                        float gm = g[(r + i) * 16 + (r + 4 + j)];     // z_p z_s^T
                        sp  += gp * gp;
                        ssq += gs * gs;
                        sm  += gm * gm;
                    }
                }
                loss += sp + ssq - 2.0f * sm;
            }
        }
        // B*N*N = 4 * 4096 * 4096
        out[0] = loss / 67108864.0f;
    }
}

// ---------------------------------------------------------------------------
extern "C" void kernel_launch(void* const* d_in, const int* in_sizes, int n_in,
                              void* d_out, int out_size, void* d_ws, size_t ws_size,
                              hipStream_t stream) {
    const float* xp = (const float*)d_in[0];   // x_pred [4,4,64,64] f32
    const float* xs = (const float*)d_in[1];   // x_src  [4,4,64,64] f32
    float* ws    = (float*)d_ws;
    float* inv   = ws;                 // 32768 floats
    float* tiles = ws + INV_ELEMS;     // 64 * 256 = 16384 floats

    cosssm_norm_kernel<<<(INV_ELEMS + 255) / 256, 256, 0, stream>>>(xp, xs, inv);
    cosssm_gram_kernel<<<NTILES, 32, 0, stream>>>(xp, xs, inv, tiles);
    cosssm_finalize_kernel<<<1, 512, 0, stream>>>(tiles, (float*)d_out);
}